// SeqSelfAttention_20349555048938
// MI455X (gfx1250) — compile-verified
//
#include <hip/hip_runtime.h>
#include <math.h>

#define BB 8
#define LL 512
#define DD 256
#define UU 32

#define SROWS 16
#define ASTRIDE 516   // 512 + 4 pad -> conflict-free b64 LDS reads in WMMA A fetch
#define CH 16         // K-rows of x staged per TDM chunk
#define XSTRIDE 264   // 256 + 8 dword TDM pad per row -> conflict-free B reads

typedef __attribute__((ext_vector_type(2))) float v2f;
typedef __attribute__((ext_vector_type(8))) float v8f;
typedef __attribute__((ext_vector_type(4))) unsigned int u32x4;
typedef __attribute__((ext_vector_type(8))) int i32x8;
typedef __attribute__((ext_vector_type(4))) int i32x4;

#if __has_builtin(__builtin_amdgcn_tanhf)
__device__ __forceinline__ float fast_tanh(float x) { return __builtin_amdgcn_tanhf(x); }
#elif __has_builtin(__builtin_amdgcn_tanh_f32)
__device__ __forceinline__ float fast_tanh(float x) { return __builtin_amdgcn_tanh_f32(x); }
#else
__device__ __forceinline__ float fast_tanh(float x) {
    float r;
    asm("v_tanh_f32 %0, %1" : "=v"(r) : "v"(x));   // native CDNA5 transcendental
    return r;
}
#endif

// ---------------------------------------------------------------------------
// TDM: DMA a [rows x 256] f32 tile (row stride 256 elems in global) into LDS,
// padding 8 dwords after every 256 dwords (LDS row stride = XSTRIDE).
// Descriptor per CDNA5 ISA 8.3/8.4. Tracked with TENSORcnt.
// ---------------------------------------------------------------------------
__device__ __forceinline__ void tdm_load_tile(unsigned lds_byte_addr,
                                              const float* gsrc, int rows) {
    unsigned long long ga = (unsigned long long)(size_t)gsrc;
    u32x4 g0;
    g0[0] = 1u;                                            // count=1, load, user
    g0[1] = lds_byte_addr;                                 // LDS dest (bytes)
    g0[2] = (unsigned)ga;                                  // global_addr[31:0]
    g0[3] = (unsigned)((ga >> 32) & 0x01FFFFFFull)         // global_addr[56:32]
            | (2u << 30);                                  // type=2 ("image")
    i32x8 g1;
    g1[0] = (2 << 16)        // data_size = 4B
          | (1 << 20)        // pad_enable
          | (7 << 22)        // pad_interval: 256 dwords
          | (7 << 25);       // pad_amount:   8 dwords
    g1[1] = (int)(256u << 16);              // tensor_dim0 = 256 (low16)
    g1[2] = (int)((unsigned)rows << 16);    // dim0 hi=0 | tensor_dim1 = rows
    g1[3] = (int)(256u << 16);              // dim1 hi=0 | tile_dim0 = 256
    g1[4] = rows;                           // tile_dim1 = rows, tile_dim2 = 0
    g1[5] = 256;                            // tensor_dim0_stride (elems) lo32
    g1[6] = 0;
    g1[7] = 0;
    i32x4 z4 = {0, 0, 0, 0};
#if __has_include(<hip/amd_detail/amd_gfx1250_TDM.h>)
    i32x8 z8 = {0, 0, 0, 0, 0, 0, 0, 0};
    __builtin_amdgcn_tensor_load_to_lds(g0, g1, z4, z4, z8, 0);   // clang-23 form
#else
    __builtin_amdgcn_tensor_load_to_lds(g0, g1, z4, z4, 0);       // ROCm 7.2 form
#endif
}

// ---------------------------------------------------------------------------
// Kernel 1: q = x@Wt + bh (which==0), k = x@Wx (which==1). Output [2][B*L][U].
// ---------------------------------------------------------------------------
__global__ void proj_kernel(const float* __restrict__ x,
                            const float* __restrict__ Wt,
                            const float* __restrict__ Wx,
                            const float* __restrict__ bh,
                            float* __restrict__ qk) {
    const int idx   = blockIdx.x * blockDim.x + threadIdx.x;  // 2*B*L*U threads
    const int u     = idx & (UU - 1);
    const int row   = (idx >> 5) & (BB * LL - 1);
    const int which = idx >> 17;                              // B*L*U == 2^17
    const float* __restrict__ W  = which ? Wx : Wt;
    const float* __restrict__ xr = x + row * DD;
    float acc = which ? 0.0f : bh[u];
#pragma unroll 8
    for (int d = 0; d < DD; ++d)
        acc = fmaf(xr[d], W[d * UU + u], acc);
    qk[idx] = acc;
}

// ---------------------------------------------------------------------------
// Kernel 2: fused tanh-attention for a tile of 16 query rows of one batch.
// ---------------------------------------------------------------------------
__launch_bounds__(256)
__global__ void attn_kernel(const float* __restrict__ x,
                            const float* __restrict__ qk,
                            const float* __restrict__ Wa,
                            const float* __restrict__ ba,
                            float* __restrict__ out) {
    __shared__ float P[SROWS * ASTRIDE];       // alpha -> softmax probs
    __shared__ float XT[2][CH * XSTRIDE];      // TDM-staged x tiles (dbl buffer)

    const int tid = threadIdx.x;
    const int b   = blockIdx.x >> 5;               // 32 s-tiles per batch
    const int s0  = (blockIdx.x & 31) * SROWS;

    const float* __restrict__ q = qk;              // bh already folded in
    const float* __restrict__ k = qk + BB * LL * UU;

    // ---------- Phase 1: alpha[s][t] = sum_u Wa[u]*tanh(qb[s][u]+k[t][u]) ----
    const int s_local = tid & (SROWS - 1);
    const float* __restrict__ qrow = q + (size_t)(b * LL + s0 + s_local) * UU;

    float qreg[UU], wreg[UU];
#pragma unroll
    for (int u4 = 0; u4 < UU / 4; ++u4) {
        float4 qa = ((const float4*)qrow)[u4];
        float4 wa = ((const float4*)Wa)[u4];
        qreg[4*u4+0] = qa.x; qreg[4*u4+1] = qa.y; qreg[4*u4+2] = qa.z; qreg[4*u4+3] = qa.w;
        wreg[4*u4+0] = wa.x; wreg[4*u4+1] = wa.y; wreg[4*u4+2] = wa.z; wreg[4*u4+3] = wa.w;
    }
    const float ba0 = ba[0];   // softmax-invariant, kept for fidelity

    const int t_first = tid >> 4;                  // 16 key-rows per pass
    for (int t = t_first; t < LL; t += 16) {
        const float4* __restrict__ krow = (const float4*)(k + (size_t)(b * LL + t) * UU);
        float acc = ba0;
#pragma unroll
        for (int u4 = 0; u4 < UU / 4; ++u4) {
            float4 kv = krow[u4];
            acc = fmaf(wreg[4*u4+0], fast_tanh(qreg[4*u4+0] + kv.x), acc);
            acc = fmaf(wreg[4*u4+1], fast_tanh(qreg[4*u4+1] + kv.y), acc);
            acc = fmaf(wreg[4*u4+2], fast_tanh(qreg[4*u4+2] + kv.z), acc);
            acc = fmaf(wreg[4*u4+3], fast_tanh(qreg[4*u4+3] + kv.w), acc);
        }
        P[s_local * ASTRIDE + t] = acc;
    }
    __syncthreads();

    // ---------- Phase 2: row softmax over t (one wave per 2 rows) ------------
    const int wave = tid >> 5;
    const int lane = tid & 31;
    for (int r = wave; r < SROWS; r += 8) {
        float* row = &P[r * ASTRIDE];
        float vals[16];
        float m = -__builtin_inff();
#pragma unroll
        for (int j = 0; j < 16; ++j) {
            vals[j] = row[lane + 32 * j];
            m = fmaxf(m, vals[j]);
        }
#pragma unroll
        for (int off = 16; off > 0; off >>= 1) m = fmaxf(m, __shfl_xor(m, off, 32));
        float ssum = 0.0f;
#pragma unroll
        for (int j = 0; j < 16; ++j) {
            vals[j] = __expf(vals[j] - m);
            ssum += vals[j];
        }
#pragma unroll
        for (int off = 16; off > 0; off >>= 1) ssum += __shfl_xor(ssum, off, 32);
        const float inv = 1.0f / ssum;
#pragma unroll
        for (int j = 0; j < 16; ++j) row[lane + 32 * j] = vals[j] * inv;
    }
    __syncthreads();

    // ---------- Phase 3: c[16,256] = P[16,512] @ x_b[512,256] ---------------
    // TDM double-buffer pipeline: wave 0 DMAs chunk c+1 while all waves do
    // WMMA on chunk c from LDS. A (16x4 f32): lanes 0-15 = M, v0={K0|K2},
    // v1={K1|K3}; B (4x16 f32): lanes 0-15 = N, v0={K0|K2}, v1={K1|K3}.
    const float* __restrict__ xb = x + (size_t)b * LL * DD;
    const int hi = lane >> 4;            // 0: lanes 0-15, 1: lanes 16-31
    const int ln = lane & 15;
    const int n0 = wave * 32;

    const unsigned xt_lds0 = (unsigned)(size_t)&XT[0][0];
    const unsigned xt_lds1 = (unsigned)(size_t)&XT[1][0];

    if (wave == 0)
        tdm_load_tile(xt_lds0, xb, CH);                       // prologue: chunk 0

    v8f acc0 = {};
    v8f acc1 = {};
    for (int c = 0; c < LL / CH; ++c) {
        if (wave == 0)
            __builtin_amdgcn_s_wait_tensorcnt(0);             // chunk c landed
        __syncthreads();                                      // visible to all
        if (wave == 0 && (c + 1) < LL / CH)                   // prefetch chunk c+1
            tdm_load_tile((c & 1) ? xt_lds0 : xt_lds1,
                          xb + (size_t)(c + 1) * CH * DD, CH);

        const float* __restrict__ Xc = &XT[c & 1][0];
        const float* __restrict__ Pa = &P[ln * ASTRIDE + c * CH];
#pragma unroll
        for (int k0l = 0; k0l < CH; k0l += 4) {
            const int kl = k0l + 2 * hi;
            v2f a = *(const v2f*)&Pa[kl];                     // ds_load_b64
            v2f b0, b1;
            b0.x = Xc[kl * XSTRIDE + n0 + ln];
            b0.y = Xc[(kl + 1) * XSTRIDE + n0 + ln];
            acc0 = __builtin_amdgcn_wmma_f32_16x16x4_f32(
                false, a, false, b0, (short)0, acc0, false, false);
            b1.x = Xc[kl * XSTRIDE + n0 + 16 + ln];
            b1.y = Xc[(kl + 1) * XSTRIDE + n0 + 16 + ln];
            acc1 = __builtin_amdgcn_wmma_f32_16x16x4_f32(
                false, a, false, b1, (short)0, acc1, false, false);
        }
        __syncthreads();                                      // done reading buf
    }

    float* __restrict__ outp = out + (size_t)(b * LL + s0) * DD + n0;
#pragma unroll
    for (int j = 0; j < 8; ++j) {
        const int m = j + hi * 8;                 // C layout: VGPR j -> M=j / j+8
        outp[m * DD + ln]      = acc0[j];
        outp[m * DD + 16 + ln] = acc1[j];
    }
}

// ---------------------------------------------------------------------------
extern "C" void kernel_launch(void* const* d_in, const int* in_sizes, int n_in,
                              void* d_out, int out_size, void* d_ws, size_t ws_size,
                              hipStream_t stream) {
    const float* x  = (const float*)d_in[0];
    const float* Wt = (const float*)d_in[1];
    const float* Wx = (const float*)d_in[2];
    const float* bh = (const float*)d_in[3];
    const float* Wa = (const float*)d_in[4];
    const float* ba = (const float*)d_in[5];
    float* out = (float*)d_out;
    float* qk  = (float*)d_ws;   // [2][B*L][U] f32 = 1 MB

    proj_kernel<<<(2 * BB * LL * UU) / 256, 256, 0, stream>>>(x, Wt, Wx, bh, qk);
    attn_kernel<<<BB * (LL / SROWS), 256, 0, stream>>>(x, qk, Wa, ba, out);
}